// EMA_Layer_12455405158674
// MI455X (gfx1250) — compile-verified
//
#include <hip/hip_runtime.h>

// EMA scan y_t = a*x_t + (1-a)*y_{t-1}, y_0 = x_0 over x[64, 8192, 64] fp32.
// Chunked scan: 16-step chunks as lower-triangular 16x16 matmuls on the CDNA5
// WMMA path (V_WMMA_F32_16X16X4_F32 chained x4 for K=16).
//
// Pass 1 streams x from HBM (x = 128 MiB -> resident in 192 MB L2),
// Pass 2 is a tiny chunk-carry scan, Pass 3 re-reads x from L2, runs the
// WMMA triangular matmuls, adds the carry term, writes y to HBM.
// HBM traffic ~= read x once + write y once = 256 MiB -> ~11.5 us floor.

#define B_DIM 64
#define T_DIM 8192
#define F_DIM 64
#define CH 16
#define NCHUNK (T_DIM / CH)   // 512
#define ALPHA_F 0.1f
#define BETA_F 0.9f

typedef float v2f __attribute__((ext_vector_type(2)));
typedef float v8f __attribute__((ext_vector_type(8)));

// exact compile-time IEEE fp32 beta^e
__host__ __device__ constexpr float bp(int e) {
    float r = 1.0f;
    for (int i = 0; i < e; ++i) r *= BETA_F;
    return r;
}
// compile-time alpha / beta^j (weight numerator for column j)
__host__ __device__ constexpr float winv(int j) { return ALPHA_F / bp(j); }

// ---------------------------------------------------------------------------
// Pass 1: per-chunk weighted tail sum (chunk's local y[15] with zero carry):
//   s[b,chunk,f] = sum_j w(j) * x[b, chunk*16+j, f]
//   w(j) = alpha*beta^(15-j); chunk 0 uses w(0) = beta^15 (y_0 = x_0 rule).
// All weights are compile-time literals -> pure FMA stream, coalesced 256B rows.
// ---------------------------------------------------------------------------
__global__ void ema_chunk_tail(const float* __restrict__ x,
                               float* __restrict__ s) {
    const int b = blockIdx.x >> 9;
    const int chunk = blockIdx.x & (NCHUNK - 1);
    const int f = threadIdx.x;            // 0..63

    const float* xb = x + ((size_t)b * T_DIM + (size_t)chunk * CH) * F_DIM + f;

    float acc = 0.0f;
#pragma unroll
    for (int j = CH - 1; j >= 1; --j) {
        acc += (ALPHA_F * bp(CH - 1 - j)) * xb[(size_t)j * F_DIM];
    }
    const float w0 = (chunk == 0) ? bp(CH - 1) : (ALPHA_F * bp(CH - 1));
    acc += w0 * xb[0];

    s[(((size_t)b << 9) + chunk) * F_DIM + f] = acc;
}

// ---------------------------------------------------------------------------
// Pass 2: chunk-level carry scan per (b, f):
//   c_0 = 0 ;  c_{k+1} = s_k + beta^16 * c_k ; overwrite s[k] with carry-IN c_k
// ---------------------------------------------------------------------------
__global__ void ema_carry_scan(float* __restrict__ s) {
    const int tid = blockIdx.x * blockDim.x + threadIdx.x;  // 0..4095
    const int b = tid >> 6;
    const int f = tid & (F_DIM - 1);

    float* p = s + ((size_t)b << 9) * F_DIM + f;

    float c = 0.0f;
#pragma unroll 8
    for (int k = 0; k < NCHUNK; ++k) {
        const float sl = p[(size_t)k * F_DIM];
        p[(size_t)k * F_DIM] = c;
        c = sl + bp(CH) * c;
    }
}

// ---------------------------------------------------------------------------
// Pass 3: per-chunk triangular matmul on WMMA + carry fix-up.
//   y[t,f] = sum_j L[t][j] x[j,f] + beta^(t+1) * c[f]
//   L[t][j] = alpha*beta^(t-j) (j<=t); chunk 0 column 0: beta^t.
// One block = one (b, chunk); 4 waves, wave w owns f-block w (16 features).
//
// Layouts (ISA 7.12.2, 32-bit, wave32):
//   A 16x4 slice : lane l(0..15)=row M=l ; vgpr0 K=2*half, vgpr1 K=2*half+1
//   B 4x16 slice : vgpr0 rows {0,2}, vgpr1 rows {1,3} striped by lane-half
//   C/D 16x16    : vgpr r -> rows {r, r+8}, N = lane%16
//
// A weights built branch-free: pl = beta^l (4 selects), then
// weight = pl * (alpha/beta^j) with alpha/beta^j a compile-time literal,
// masked by (l >= j). No divergent regions around the WMMAs; EXEC stays all-1.
// ---------------------------------------------------------------------------
__global__ void ema_chunk_wmma(const float* __restrict__ x,
                               const float* __restrict__ carry,
                               float* __restrict__ y) {
    const int b = blockIdx.x >> 9;
    const int chunk = blockIdx.x & (NCHUNK - 1);
    const int wave = threadIdx.x >> 5;    // 0..3 -> f-block
    const int lane = threadIdx.x & 31;
    const int half = lane >> 4;
    const int l = lane & 15;

    const int tb = chunk * CH;
    const int fb = wave * 16;

    const float* xb = x + ((size_t)b * T_DIM + tb) * F_DIM;
    float* yb = y + ((size_t)b * T_DIM + tb) * F_DIM;

    // pl = beta^l, straight-line (4 cndmask + 3 mul)
    const float pl = ((l & 1) ? bp(1) : 1.0f) *
                     ((l & 2) ? bp(2) : 1.0f) *
                     ((l & 4) ? bp(4) : 1.0f) *
                     ((l & 8) ? bp(8) : 1.0f);

    // chunk-0 column-0 scale: 1.0 (y_0 = x_0) instead of alpha. Uniform.
    const float col0 = (chunk == 0) ? 1.0f : ALPHA_F;

    v8f acc = {};
#pragma unroll
    for (int s = 0; s < 4; ++s) {
        const int j0 = 4 * s + 2 * half;
        const int j1 = j0 + 1;

        // per-slice column scales: compile-time literals selected by half
        const float sc0 = half ? winv(4 * s + 2)
                               : (s == 0 ? col0 : winv(4 * s));
        const float sc1 = half ? winv(4 * s + 3) : winv(4 * s + 1);

        v2f a;
        a.x = (l >= j0) ? sc0 * pl : 0.0f;
        a.y = (l >= j1) ? sc1 * pl : 0.0f;

        v2f bm;
        bm.x = xb[(size_t)j0 * F_DIM + fb + l];
        bm.y = xb[(size_t)j1 * F_DIM + fb + l];

        acc = __builtin_amdgcn_wmma_f32_16x16x4_f32(
            /*neg_a=*/false, a, /*neg_b=*/false, bm,
            /*c_mod=*/(short)0, acc, /*reuse_a=*/false, /*reuse_b=*/false);
    }

    // carry fix-up + store: D vgpr r covers rows t = r + 8*half,
    // carry factor beta^(t+1) is a 2-literal select per r.
    const float cv = carry[(((size_t)b << 9) + chunk) * F_DIM + fb + l];
#pragma unroll
    for (int r = 0; r < 8; ++r) {
        const float cf = half ? bp(r + 9) : bp(r + 1);
        const int t = r + 8 * half;
        yb[(size_t)t * F_DIM + fb + l] = acc[r] + cf * cv;
    }
}

extern "C" void kernel_launch(void* const* d_in, const int* in_sizes, int n_in,
                              void* d_out, int out_size, void* d_ws, size_t ws_size,
                              hipStream_t stream) {
    (void)in_sizes; (void)n_in; (void)out_size; (void)ws_size;

    const float* x = (const float*)d_in[0];
    float* out = (float*)d_out;
    float* s = (float*)d_ws;   // 64*512*64*4 = 8 MiB scratch

    ema_chunk_tail<<<dim3(B_DIM * NCHUNK), dim3(F_DIM), 0, stream>>>(x, s);
    ema_carry_scan<<<dim3(32), dim3(128), 0, stream>>>(s);
    ema_chunk_wmma<<<dim3(B_DIM * NCHUNK), dim3(128), 0, stream>>>(x, s, out);
}